// AttentionLayer_43404939493525
// MI455X (gfx1250) — compile-verified
//
#include <hip/hip_runtime.h>
#include <math.h>

#define B_ 256
#define W_ 512
#define F_ 512
#define N_ 512

typedef __attribute__((ext_vector_type(2))) float v2f;
typedef __attribute__((ext_vector_type(8))) float v8f;

// ---------------------------------------------------------------------------
// Kernel 1: t[b,f] = sum_w x[b,w,f] * wvec[w]
// One block per b, 512 threads (one per f). Coalesced 2KB row reads.
// ---------------------------------------------------------------------------
__global__ __launch_bounds__(512) void k_reduce_w(const float* __restrict__ x,
                                                  const float* __restrict__ wvec,
                                                  float* __restrict__ t) {
    __shared__ float ws[W_];
    const int b = blockIdx.x;
    const int f = threadIdx.x;
    ws[f] = wvec[f];
    __syncthreads();
    const float* xb = x + (size_t)b * W_ * F_ + f;
    float acc = 0.f;
#pragma unroll 8
    for (int w = 0; w < W_; ++w) {
        if ((w & 63) == 0) __builtin_prefetch(xb + (size_t)(w + 64) * F_, 0, 1);
        acc = __builtin_fmaf(xb[(size_t)w * F_], ws[w], acc);
    }
    t[b * F_ + f] = acc;
}

// ---------------------------------------------------------------------------
// WMMA fp32 GEMM: D(MxN) = A(MxK) @ B(KxN)   (TRANSB: B element (k,n) = Bm[n*K+k])
// One wave per 16x16 tile of D, K-loop in steps of 4 using V_WMMA_F32_16X16X4_F32.
// A 16x4 layout: lane m = lane%16, holds K = {kh, kh+1}, kh = 2*(lane/16).
// B 4x16 layout: lane n = lane%16, holds K = {kh, kh+1}.
// C/D: VGPR r holds row r + 8*(lane/16), col = lane%16.
// All branches are wave-uniform -> EXEC all-1s for WMMA.
// ---------------------------------------------------------------------------
template <bool TRANSB>
__global__ __launch_bounds__(128) void k_gemm_wmma(const float* __restrict__ A,
                                                   const float* __restrict__ Bm,
                                                   float* __restrict__ D,
                                                   int M, int N, int K) {
    const int wave = threadIdx.x >> 5;
    const int lane = threadIdx.x & 31;
    const int tilesN = N >> 4;
    const int tile = blockIdx.x * 4 + wave;
    const int tm = tile / tilesN;
    const int tn = tile - tm * tilesN;
    if (tm * 16 >= M) return;  // wave-uniform

    const int nm   = lane & 15;          // m for A, n for B/C/D
    const int kh   = (lane >> 4) << 1;   // 0 or 2
    const float* arow = A + (size_t)(tm * 16 + nm) * K;

    v8f acc = {0.f, 0.f, 0.f, 0.f, 0.f, 0.f, 0.f, 0.f};
    for (int k0 = 0; k0 < K; k0 += 4) {
        v2f a, b;
        a.x = arow[k0 + kh];
        a.y = arow[k0 + kh + 1];
        if (TRANSB) {
            const float* brow = Bm + (size_t)(tn * 16 + nm) * K;
            b.x = brow[k0 + kh];
            b.y = brow[k0 + kh + 1];
        } else {
            b.x = Bm[(size_t)(k0 + kh) * N + tn * 16 + nm];
            b.y = Bm[(size_t)(k0 + kh + 1) * N + tn * 16 + nm];
        }
        acc = __builtin_amdgcn_wmma_f32_16x16x4_f32(
            /*neg_a=*/false, a, /*neg_b=*/false, b,
            /*c_mod=*/(short)0, acc, /*reuse_a=*/false, /*reuse_b=*/false);
    }
    const int rbase = (lane >> 4) << 3;
#pragma unroll
    for (int r = 0; r < 8; ++r)
        D[(size_t)(tm * 16 + rbase + r) * N + tn * 16 + nm] = acc[r];
}

// ---------------------------------------------------------------------------
// Kernel 3: per-b fused  scores = (x_b @ Qk_b)/sqrt(N) -> softmax -> c_b = w^T x_b
// One block per b, 512 threads (16 waves). x_b (1MB) read twice, L2-resident.
// ---------------------------------------------------------------------------
__global__ __launch_bounds__(512) void k_attn(const float* __restrict__ x,
                                              const float* __restrict__ Qk,
                                              float* __restrict__ c) {
    __shared__ float qk_s[F_];
    __shared__ float sw[W_];
    __shared__ float red[16];
    const int b    = blockIdx.x;
    const int tid  = threadIdx.x;
    const int lane = tid & 31;
    const int wave = tid >> 5;
    const float scale = 0.044194173824159216f;  // 1/sqrt(512)

    qk_s[tid] = Qk[b * F_ + tid];
    __syncthreads();

    const float* xb = x + (size_t)b * W_ * F_;

    // ---- scores: each wave owns rows w = wave, wave+16, ... (32 rows/wave)
    for (int w = wave; w < W_; w += 16) {
        const float* row = xb + (size_t)w * F_;
        __builtin_prefetch(row + 16 * F_, 0, 1);
        float p = 0.f;
#pragma unroll
        for (int f = lane; f < F_; f += 32) p = __builtin_fmaf(row[f], qk_s[f], p);
#pragma unroll
        for (int off = 16; off > 0; off >>= 1) p += __shfl_down(p, off, 32);
        if (lane == 0) sw[w] = p;
    }
    __syncthreads();

    // ---- softmax over sw[0..511]
    float s = sw[tid] * scale;
    float wm = s;
#pragma unroll
    for (int off = 16; off > 0; off >>= 1) wm = fmaxf(wm, __shfl_xor(wm, off, 32));
    if (lane == 0) red[wave] = wm;
    __syncthreads();
    if (tid < 32) {
        float v = (lane < 16) ? red[lane] : -3.4e38f;
#pragma unroll
        for (int off = 8; off > 0; off >>= 1) v = fmaxf(v, __shfl_xor(v, off, 32));
        if (lane == 0) red[0] = v;
    }
    __syncthreads();
    const float m = red[0];
    const float e = __expf(s - m);
    float se = e;
#pragma unroll
    for (int off = 16; off > 0; off >>= 1) se += __shfl_xor(se, off, 32);
    __syncthreads();  // everyone has read red[0]
    if (lane == 0) red[wave] = se;
    __syncthreads();
    if (tid < 32) {
        float v = (lane < 16) ? red[lane] : 0.f;
#pragma unroll
        for (int off = 8; off > 0; off >>= 1) v += __shfl_xor(v, off, 32);
        if (lane == 0) red[0] = v;
    }
    __syncthreads();
    const float invZ = 1.f / red[0];
    sw[tid] = e * invZ;
    __syncthreads();

    // ---- context: c[b,f] = sum_w sw[w] * x[b,w,f]  (thread per f, coalesced)
    float acc = 0.f;
#pragma unroll 8
    for (int w = 0; w < W_; ++w) {
        if ((w & 63) == 0) __builtin_prefetch(xb + (size_t)(w + 64) * F_ + tid, 0, 1);
        acc = __builtin_fmaf(xb[(size_t)w * F_ + tid], sw[w], acc);
    }
    c[b * F_ + tid] = acc;
}

// ---------------------------------------------------------------------------
extern "C" void kernel_launch(void* const* d_in, const int* in_sizes, int n_in,
                              void* d_out, int out_size, void* d_ws, size_t ws_size,
                              hipStream_t stream) {
    (void)in_sizes; (void)n_in; (void)out_size; (void)ws_size;
    const float* x    = (const float*)d_in[0];  // (B, W, F)
    const float* wk   = (const float*)d_in[1];  // (F, N)
    const float* wq   = (const float*)d_in[2];  // (F, N)
    const float* wv   = (const float*)d_in[3];  // (F, N)
    const float* wvec = (const float*)d_in[4];  // (W, 1)
    float* out = (float*)d_out;                 // (B, N)

    float* t  = (float*)d_ws;        // (B, F)  512 KB
    float* Q  = t  + B_ * F_;        // (B, N)  512 KB
    float* Qk = Q  + B_ * N_;        // (B, F)  512 KB
    float* c  = t;                   // reuse t (dead after Q is computed)

    // 1) t = einsum("bwf,w->bf")
    k_reduce_w<<<B_, 512, 0, stream>>>(x, wvec, t);

    // 2) Q = t @ Wq        (M=256, N=512, K=512)   -> 512 tiles / 4 waves = 128 blocks
    k_gemm_wmma<false><<<128, 128, 0, stream>>>(t, wq, Q, B_, N_, F_);

    // 3) Qk = Q @ Wk^T     (M=256, N=F=512, K=N=512)
    k_gemm_wmma<true><<<128, 128, 0, stream>>>(Q, wk, Qk, B_, F_, N_);

    // 4) scores -> softmax -> context c
    k_attn<<<B_, 512, 0, stream>>>(x, Qk, c);

    // 5) out = c @ Wv
    k_gemm_wmma<false><<<128, 128, 0, stream>>>(c, wv, out, B_, N_, F_);
}